// MemoryAccess_70566312673534
// MI455X (gfx1250) — compile-verified
//
#include <hip/hip_runtime.h>
#include <hip/hip_bf16.h>
#include <math.h>

// Problem constants (match reference)
constexpr int B  = 8;
constexpr int M  = 2048;
constexpr int W  = 64;
constexpr int NW = 2;
constexpr int NR = 4;
constexpr int H  = 4;
constexpr float EPS = 1e-6f;

// Output sizes / offsets (flat concatenation in return order)
constexpr size_t N_MEM   = (size_t)B * M * W;       // updated_memory
constexpr size_t N_COS   = (size_t)B * H * M;       // cosine_output
constexpr size_t N_LINK  = (size_t)B * NW * M * M;  // new_link
constexpr size_t N_PREC  = (size_t)B * NW * M;      // new_precedence
constexpr size_t N_USE   = (size_t)B * M;           // updated_usage
// read_output = N_MEM

typedef __attribute__((ext_vector_type(2))) float v2f;
typedef __attribute__((ext_vector_type(4))) float v4f;
typedef __attribute__((ext_vector_type(8))) float v8f;

// ---------------------------------------------------------------------------
// Kernel 1: link update (dominant, pure HBM streaming, non-temporal b128)
// new_link[b,n,i,j] = (1 - w_i - w_j)*prev_link + w_i*prev_prec_j ; diag = 0
// one block per (b,n,i) row of 2048 floats
// ---------------------------------------------------------------------------
__global__ __launch_bounds__(256) void link_kernel(
    const float* __restrict__ ww,         // B,NW,M
    const float* __restrict__ prev_prec,  // B,NW,M
    const float* __restrict__ prev_link,  // B,NW,M,M
    float* __restrict__ out_link)         // B,NW,M,M
{
    const int row = blockIdx.x;          // [0, B*NW*M)
    const int i   = row & (M - 1);
    const int bn  = row >> 11;           // M = 2048 = 2^11
    const float wi = ww[bn * M + i];     // uniform -> scalar load

    const v4f* __restrict__ wj4 = (const v4f*)(ww + (size_t)bn * M);
    const v4f* __restrict__ pj4 = (const v4f*)(prev_prec + (size_t)bn * M);
    const v4f* __restrict__ L4  = (const v4f*)(prev_link + (size_t)row * M);
    v4f* __restrict__       O4  = (v4f*)(out_link + (size_t)row * M);

#pragma unroll
    for (int t = 0; t < 2; ++t) {
        const int j4 = threadIdx.x + t * 256;   // float4 index; j = 4*j4
        const v4f wj = wj4[j4];                 // cached (L2-resident rows)
        const v4f pj = pj4[j4];
        const v4f L  = __builtin_nontemporal_load(L4 + j4);  // streaming
        v4f o;
        o.x = (1.f - wi - wj.x) * L.x + wi * pj.x;
        o.y = (1.f - wi - wj.y) * L.y + wi * pj.y;
        o.z = (1.f - wi - wj.z) * L.z + wi * pj.z;
        o.w = (1.f - wi - wj.w) * L.w + wi * pj.w;
        const int jb = j4 * 4;
        if (i >= jb && i < jb + 4) o[i - jb] = 0.f;  // zero diagonal
        __builtin_nontemporal_store(o, O4 + j4);
    }
}

// ---------------------------------------------------------------------------
// Kernel 2: cosine similarity + softmax. One block per (b,h).
// ---------------------------------------------------------------------------
__global__ __launch_bounds__(256) void cosine_kernel(
    const float* __restrict__ memory,     // B,M,W
    const float* __restrict__ keys,       // B,H,W
    const float* __restrict__ strengths,  // B,H
    float* __restrict__ out_cos)          // B,H,M
{
    const int bh = blockIdx.x;            // [0, B*H)
    const int tid = threadIdx.x;
    __shared__ float kcache[W];
    __shared__ float red[256];

    if (tid < W) kcache[tid] = keys[(size_t)bh * W + tid];
    __syncthreads();

    float kn = 0.f;
#pragma unroll
    for (int w = 0; w < W; ++w) { float k = kcache[w]; kn += k * k; }
    kn = sqrtf(kn);

    const float s = strengths[bh];
    const float beta = (s > 20.f) ? s : log1pf(expf(s));   // softplus

    const int b = bh / H;
    float simv[8];
    float lmax = -3.4e38f;
#pragma unroll
    for (int t = 0; t < 8; ++t) {
        const int m = tid + t * 256;
        const float* __restrict__ rowp = memory + ((size_t)b * M + m) * W;
        float dot = 0.f, mn = 0.f;
#pragma unroll
        for (int w = 0; w < W; ++w) {
            const float mv = rowp[w];
            dot += mv * kcache[w];
            mn  += mv * mv;
        }
        mn = sqrtf(mn);
        simv[t] = dot / (kn * mn + EPS) * beta;
        lmax = fmaxf(lmax, simv[t]);
    }
    // block-reduce max
    red[tid] = lmax; __syncthreads();
    for (int st = 128; st > 0; st >>= 1) {
        if (tid < st) red[tid] = fmaxf(red[tid], red[tid + st]);
        __syncthreads();
    }
    const float gmax = red[0]; __syncthreads();
    // exp + block-reduce sum
    float lsum = 0.f;
#pragma unroll
    for (int t = 0; t < 8; ++t) { simv[t] = expf(simv[t] - gmax); lsum += simv[t]; }
    red[tid] = lsum; __syncthreads();
    for (int st = 128; st > 0; st >>= 1) {
        if (tid < st) red[tid] += red[tid + st];
        __syncthreads();
    }
    const float inv = 1.f / red[0];
#pragma unroll
    for (int t = 0; t < 8; ++t) {
        const int m = tid + t * 256;
        out_cos[(size_t)bh * M + m] = simv[t] * inv;
    }
}

// ---------------------------------------------------------------------------
// Kernel 3: per-(b,n) sum of write weights -> ws  (16 scalars)
// ---------------------------------------------------------------------------
__global__ __launch_bounds__(256) void wsum_kernel(
    const float* __restrict__ ww, float* __restrict__ wsum)
{
    const int bn = blockIdx.x;            // [0, B*NW)
    const int tid = threadIdx.x;
    __shared__ float red[256];
    float s = 0.f;
    for (int m = tid; m < M; m += 256) s += ww[(size_t)bn * M + m];
    red[tid] = s; __syncthreads();
    for (int st = 128; st > 0; st >>= 1) {
        if (tid < st) red[tid] += red[tid + st];
        __syncthreads();
    }
    if (tid == 0) wsum[bn] = red[0];
}

// ---------------------------------------------------------------------------
// Kernel 4: new_precedence + updated_usage  (elementwise over B*M)
// ---------------------------------------------------------------------------
__global__ __launch_bounds__(256) void prec_usage_kernel(
    const float* __restrict__ ww,         // B,NW,M
    const float* __restrict__ prev_prec,  // B,NW,M
    const float* __restrict__ prev_usage, // B,M
    const float* __restrict__ free_gate,  // B,NR
    const float* __restrict__ rw,         // B,NR,M
    const float* __restrict__ wsum,       // B*NW
    float* __restrict__ out_prec,         // B,NW,M
    float* __restrict__ out_usage)        // B,M
{
    const int idx = blockIdx.x * 256 + threadIdx.x;  // b*M + m
    const int b = idx / M;
    const int m = idx - b * M;

    float agg = 1.f;
#pragma unroll
    for (int n = 0; n < NW; ++n) {
        const size_t o = ((size_t)b * NW + n) * M + m;
        const float wv = ww[o];
        out_prec[o] = (1.f - wsum[b * NW + n]) * prev_prec[o] + wv;
        agg *= (1.f - wv);
    }
    agg = 1.f - agg;
    const float pu = prev_usage[idx];
    const float uaw = pu + (1.f - pu) * agg;
    float ret = 1.f;
#pragma unroll
    for (int r = 0; r < NR; ++r) {
        ret *= (1.f - free_gate[b * NR + r] * rw[((size_t)b * NR + r) * M + m]);
    }
    out_usage[idx] = uaw * ret;
}

// ---------------------------------------------------------------------------
// Kernel 5: fused memory update + read output via V_WMMA_F32_16X16X4_F32.
// D[m,w] = sum_k A[m,k]*B[k,w]; erase/write: K=NW=2 (padded to 4), read: K=NR=4.
// One wave per 16-row m-tile; loops over 4 w-tiles of 16.
// A layout (16x4 f32): vgpr v, lane-half h -> element (m=lane%16, k=2h+v)
// B layout (4x16 f32): vgpr v, lane-half h -> element (k=2h+v, w=lane%16)
// D layout (16x16 f32): vgpr r -> (m = r + 8h, w = lane%16)
// ---------------------------------------------------------------------------
__global__ __launch_bounds__(256) void memupd_kernel(
    const float* __restrict__ memory,  // B,M,W
    const float* __restrict__ ww,      // B,NW,M
    const float* __restrict__ ev,      // B,NW,W
    const float* __restrict__ wv,      // B,NW,W
    const float* __restrict__ rw,      // B,NR,M
    const float* __restrict__ rv,      // B,NR,W
    float* __restrict__ out_mem,       // B,M,W
    float* __restrict__ out_read)      // B,M,W
{
    const int lane = threadIdx.x & 31;
    const int wave = threadIdx.x >> 5;                 // 0..7
    const int tile = blockIdx.x * 8 + wave;            // [0, B*M/16)
    const int b    = tile / (M / 16);
    const int m0   = (tile - b * (M / 16)) * 16;
    const int half = lane >> 4;
    const int lo   = lane & 15;
    const int m    = m0 + lo;

    // A for erase/write: k = 2*half + v, valid only k < NW=2 (half==0)
    v2f a_w;
    a_w.x = (half == 0) ? ww[((size_t)b * NW + 0) * M + m] : 0.f;
    a_w.y = (half == 0) ? ww[((size_t)b * NW + 1) * M + m] : 0.f;
    // A for read: k = 2*half + v, NR=4 exactly fills K
    v2f a_r;
    a_r.x = rw[((size_t)b * NR + 2 * half + 0) * M + m];
    a_r.y = rw[((size_t)b * NR + 2 * half + 1) * M + m];

#pragma unroll
    for (int wt = 0; wt < 4; ++wt) {
        const int wc = wt * 16 + lo;
        v2f b_e, b_w, b_r;
        b_e.x = (half == 0) ? ev[((size_t)b * NW + 0) * W + wc] : 0.f;
        b_e.y = (half == 0) ? ev[((size_t)b * NW + 1) * W + wc] : 0.f;
        b_w.x = (half == 0) ? wv[((size_t)b * NW + 0) * W + wc] : 0.f;
        b_w.y = (half == 0) ? wv[((size_t)b * NW + 1) * W + wc] : 0.f;
        b_r.x = rv[((size_t)b * NR + 2 * half + 0) * W + wc];
        b_r.y = rv[((size_t)b * NR + 2 * half + 1) * W + wc];

        v8f z = {};
        v8f d_e = __builtin_amdgcn_wmma_f32_16x16x4_f32(
            false, a_w, false, b_e, (short)0, z, false, false);
        v8f d_w = __builtin_amdgcn_wmma_f32_16x16x4_f32(
            false, a_w, false, b_w, (short)0, z, false, false);
        v8f d_r = __builtin_amdgcn_wmma_f32_16x16x4_f32(
            false, a_r, false, b_r, (short)0, z, false, false);

#pragma unroll
        for (int r = 0; r < 8; ++r) {
            const int mm = m0 + r + 8 * half;
            const size_t idx = ((size_t)b * M + mm) * W + wc;
            float e = d_e[r];
            e = fminf(fmaxf(e, 0.f), 1.f);                 // clip(erase,0,1)
            out_mem[idx]  = memory[idx] * (1.f - e) + d_w[r];
            out_read[idx] = d_r[r];
        }
    }
}

// ---------------------------------------------------------------------------
extern "C" void kernel_launch(void* const* d_in, const int* in_sizes, int n_in,
                              void* d_out, int out_size, void* d_ws, size_t ws_size,
                              hipStream_t stream) {
    const float* memory     = (const float*)d_in[0];
    const float* keys       = (const float*)d_in[1];
    const float* strengths  = (const float*)d_in[2];
    const float* wwts       = (const float*)d_in[3];
    const float* free_gate  = (const float*)d_in[4];
    const float* rwts       = (const float*)d_in[5];
    const float* prev_link  = (const float*)d_in[6];
    const float* prev_prec  = (const float*)d_in[7];
    const float* prev_usage = (const float*)d_in[8];
    const float* erase_vec  = (const float*)d_in[9];
    const float* write_vec  = (const float*)d_in[10];
    const float* read_vec   = (const float*)d_in[11];

    float* out = (float*)d_out;
    float* out_mem   = out;
    float* out_cos   = out_mem + N_MEM;
    float* out_link  = out_cos + N_COS;
    float* out_prec  = out_link + N_LINK;
    float* out_usage = out_prec + N_PREC;
    float* out_read  = out_usage + N_USE;

    float* wsum = (float*)d_ws;   // B*NW floats of scratch

    // small kernels first (all independent except wsum -> prec_usage)
    cosine_kernel<<<B * H, 256, 0, stream>>>(memory, keys, strengths, out_cos);
    wsum_kernel<<<B * NW, 256, 0, stream>>>(wwts, wsum);
    prec_usage_kernel<<<(B * M) / 256, 256, 0, stream>>>(
        wwts, prev_prec, prev_usage, free_gate, rwts, wsum, out_prec, out_usage);
    memupd_kernel<<<(B * M / 16) / 8, 256, 0, stream>>>(
        memory, wwts, erase_vec, write_vec, rwts, read_vec, out_mem, out_read);
    // dominant streaming kernel last
    link_kernel<<<B * NW * M, 256, 0, stream>>>(wwts, prev_prec, prev_link, out_link);
}